// WeightNet_3453153706430
// MI455X (gfx1250) — compile-verified
//
#include <hip/hip_runtime.h>
#include <hip/hip_bf16.h>

// ---------------------------------------------------------------------------
// Per-point expert-indexed MLP (7 -> 16 -> 32 -> 16, LeakyReLU 0.2), L=16
// Strategy: bin points by expert, then WMMA fp32 16x16x4 tiles of 16
// same-expert points. Memory-bound target ~48MB @ 23.3 TB/s.
// ---------------------------------------------------------------------------

typedef float v2f __attribute__((ext_vector_type(2)));
typedef float v8f __attribute__((ext_vector_type(8)));

#define NEG_SLOPE 0.2f
#define LRELU(v) ((v) > 0.0f ? (v) : NEG_SLOPE * (v))

// ---------------- Pass 1: bucket points by expert index -------------------

__global__ void zero_header_kernel(int* hdr) {
    if (threadIdx.x < 48) hdr[threadIdx.x] = 0;
}

__global__ void hist_kernel(const int* __restrict__ idx, int n, int* __restrict__ counts) {
    __shared__ int h[16];
    if (threadIdx.x < 16) h[threadIdx.x] = 0;
    __syncthreads();
    int i = blockIdx.x * blockDim.x + threadIdx.x;
    if (i < n) atomicAdd(&h[idx[i] & 15], 1);
    __syncthreads();
    if (threadIdx.x < 16 && h[threadIdx.x] > 0)
        atomicAdd(&counts[threadIdx.x], h[threadIdx.x]);
}

__global__ void scan_kernel(const int* __restrict__ counts, int* __restrict__ offs,
                            int* __restrict__ cursors) {
    if (threadIdx.x == 0) {
        int s = 0;
        for (int e = 0; e < 16; ++e) {
            offs[e] = s;
            cursors[e] = s;
            s += counts[e];
        }
    }
}

__global__ void scatter_kernel(const int* __restrict__ idx, int n,
                               int* __restrict__ cursors, int* __restrict__ perm) {
    __shared__ int lcount[16];
    __shared__ int lbase[16];
    if (threadIdx.x < 16) lcount[threadIdx.x] = 0;
    __syncthreads();
    int i = blockIdx.x * blockDim.x + threadIdx.x;
    int e = 0, rank = 0;
    bool live = (i < n);
    if (live) {
        e = idx[i] & 15;
        rank = atomicAdd(&lcount[e], 1);
    }
    __syncthreads();
    if (threadIdx.x < 16 && lcount[threadIdx.x] > 0)
        lbase[threadIdx.x] = atomicAdd(&cursors[threadIdx.x], lcount[threadIdx.x]);
    __syncthreads();
    if (live) perm[lbase[e] + rank] = i;
}

// ---------------- Pass 2: WMMA expert tiles --------------------------------
//
// One wave (32 lanes) per block. blockIdx.x = expert, blockIdx.y strides
// over this expert's 16-point tiles.
//
// v_wmma_f32_16x16x4_f32 layouts (wave32):
//   A (16x4): lane L<16 holds row M=L, {K=0,K=1}; lane L+16 holds {K=2,K=3}
//   B (4x16): lane L<16 holds col N=L, {K=0,K=1}; lane L+16 holds {K=2,K=3}
//   C (16x16): VGPR r, lanes 0-15 = (M=r, N=lane); lanes 16-31 = (M=r+8, N=lane-16)

__global__ __launch_bounds__(32) void expert_mlp_wmma_kernel(
    const float* __restrict__ x,
    const float* __restrict__ W0, const float* __restrict__ b0,
    const float* __restrict__ W1, const float* __restrict__ b1,
    const float* __restrict__ W2, const float* __restrict__ b2,
    const int* __restrict__ offs, const int* __restrict__ counts,
    const int* __restrict__ perm,
    float* __restrict__ out)
{
    const int e    = blockIdx.x;
    const int lane = threadIdx.x;
    const int m    = lane & 15;
    const int hi   = lane >> 4;

    __shared__ float sW0[8 * 16];   // K padded 7 -> 8 (row 7 zeroed)
    __shared__ float sW1[16 * 32];
    __shared__ float sW2[32 * 16];
    __shared__ float sB0[16], sB1[32], sB2[16];
    __shared__ float sX[16 * 8];    // input tile, K padded 7 -> 8
    __shared__ float sH[16 * 33];   // activations, stride 33 (bank-conflict-free)
    __shared__ int   sRow[16];

    // ---- stage expert weights into LDS (once per block) ----
    for (int i = lane; i < 128; i += 32) {
        int k = i >> 4, n = i & 15;
        sW0[i] = (k < 7) ? W0[e * 112 + k * 16 + n] : 0.0f;
    }
    for (int i = lane; i < 512; i += 32) sW1[i] = W1[e * 512 + i];
    for (int i = lane; i < 512; i += 32) sW2[i] = W2[e * 512 + i];
    if (lane < 16) sB0[lane] = b0[e * 16 + lane];
    sB1[lane] = b1[e * 32 + lane];
    if (lane < 16) sB2[lane] = b2[e * 16 + lane];
    __syncthreads();

    // ---- hoist all B fragments into registers (reused across tiles) ----
    v2f bf0[2], bf1[2][4], bf2[8];
#pragma unroll
    for (int c = 0; c < 2; ++c) {
        int kk = 4 * c + 2 * hi;
        bf0[c].x = sW0[kk * 16 + m];
        bf0[c].y = sW0[(kk + 1) * 16 + m];
    }
#pragma unroll
    for (int j = 0; j < 2; ++j)
#pragma unroll
        for (int c = 0; c < 4; ++c) {
            int kk = 4 * c + 2 * hi;
            int n  = 16 * j + m;
            bf1[j][c].x = sW1[kk * 32 + n];
            bf1[j][c].y = sW1[(kk + 1) * 32 + n];
        }
#pragma unroll
    for (int c = 0; c < 8; ++c) {
        int kk = 4 * c + 2 * hi;
        bf2[c].x = sW2[kk * 16 + m];
        bf2[c].y = sW2[(kk + 1) * 16 + m];
    }

    const float bias0 = sB0[m];
    const float b1lo  = sB1[m];
    const float b1hi  = sB1[16 + m];
    const float bias2 = sB2[m];

    const int base   = offs[e];
    const int cnt    = counts[e];
    const int ntiles = (cnt + 15) >> 4;

    for (int t = blockIdx.y; t < ntiles; t += gridDim.y) {
        __syncthreads();  // protect sX / sRow / sH from previous iteration readers

        int valid = cnt - (t << 4);
        if (valid > 16) valid = 16;

        // ---- gather 16 points of x into LDS ----
        if (lane < 16) {
            int row = (lane < valid) ? perm[base + (t << 4) + lane] : -1;
            sRow[lane] = row;
            if (row >= 0) {
                const float* xs = x + row * 7;
#pragma unroll
                for (int k = 0; k < 7; ++k) sX[lane * 8 + k] = xs[k];
            } else {
#pragma unroll
                for (int k = 0; k < 7; ++k) sX[lane * 8 + k] = 0.0f;
            }
            sX[lane * 8 + 7] = 0.0f;  // K-pad
        }
        __syncthreads();

        // ---- layer 0: [16x8] @ [8x16] ----
        v2f a0, a1;
        a0.x = sX[m * 8 + 2 * hi];
        a0.y = sX[m * 8 + 2 * hi + 1];
        a1.x = sX[m * 8 + 4 + 2 * hi];
        a1.y = sX[m * 8 + 4 + 2 * hi + 1];
        v8f c0 = {};
        c0 = __builtin_amdgcn_wmma_f32_16x16x4_f32(false, a0, false, bf0[0], (short)0, c0, false, false);
        c0 = __builtin_amdgcn_wmma_f32_16x16x4_f32(false, a1, false, bf0[1], (short)0, c0, false, false);
#pragma unroll
        for (int r = 0; r < 8; ++r) {
            float v = c0[r] + bias0;
            c0[r] = LRELU(v);
        }
        // store H0 (16x16) to LDS
#pragma unroll
        for (int r = 0; r < 8; ++r) sH[(r + 8 * hi) * 33 + m] = c0[r];
        __syncthreads();

        // ---- layer 1: [16x16] @ [16x32] -> two 16x16 N-tiles ----
        v2f af1[4];
#pragma unroll
        for (int c = 0; c < 4; ++c) {
            int kk = 4 * c + 2 * hi;
            af1[c].x = sH[m * 33 + kk];
            af1[c].y = sH[m * 33 + kk + 1];
        }
        __syncthreads();  // A frags in regs; safe to overwrite sH below

        v8f c1lo = {}, c1hi = {};
#pragma unroll
        for (int c = 0; c < 4; ++c)
            c1lo = __builtin_amdgcn_wmma_f32_16x16x4_f32(false, af1[c], false, bf1[0][c], (short)0, c1lo, false, false);
#pragma unroll
        for (int c = 0; c < 4; ++c)
            c1hi = __builtin_amdgcn_wmma_f32_16x16x4_f32(false, af1[c], false, bf1[1][c], (short)0, c1hi, false, false);
#pragma unroll
        for (int r = 0; r < 8; ++r) {
            float vlo = c1lo[r] + b1lo;
            float vhi = c1hi[r] + b1hi;
            sH[(r + 8 * hi) * 33 + m]      = LRELU(vlo);
            sH[(r + 8 * hi) * 33 + 16 + m] = LRELU(vhi);
        }
        __syncthreads();

        // ---- layer 2: [16x32] @ [32x16] ----
        v8f c2 = {};
#pragma unroll
        for (int c = 0; c < 8; ++c) {
            int kk = 4 * c + 2 * hi;
            v2f a;
            a.x = sH[m * 33 + kk];
            a.y = sH[m * 33 + kk + 1];
            c2 = __builtin_amdgcn_wmma_f32_16x16x4_f32(false, a, false, bf2[c], (short)0, c2, false, false);
        }

        // ---- bias + LeakyReLU + scatter-store to original rows ----
#pragma unroll
        for (int r = 0; r < 8; ++r) {
            int row = sRow[r + 8 * hi];
            if (row >= 0) {
                float v = c2[r] + bias2;
                out[row * 16 + m] = LRELU(v);
            }
        }
    }
}

// ---------------------------------------------------------------------------

extern "C" void kernel_launch(void* const* d_in, const int* in_sizes, int n_in,
                              void* d_out, int out_size, void* d_ws, size_t ws_size,
                              hipStream_t stream) {
    const float* x  = (const float*)d_in[0];
    const int*   idx = (const int*)d_in[1];
    const float* W0 = (const float*)d_in[2];
    const float* b0 = (const float*)d_in[3];
    const float* W1 = (const float*)d_in[4];
    const float* b1 = (const float*)d_in[5];
    const float* W2 = (const float*)d_in[6];
    const float* b2 = (const float*)d_in[7];
    float* out = (float*)d_out;

    const int n = in_sizes[1];  // number of points (idx count)

    // workspace layout: [0..15] counts | [16..31] offsets | [32..47] cursors
    //                   | [64 ..] perm (n ints)
    int* hdr     = (int*)d_ws;
    int* counts  = hdr;
    int* offs    = hdr + 16;
    int* cursors = hdr + 32;
    int* perm    = hdr + 64;

    const int nb = (n + 255) / 256;

    zero_header_kernel<<<1, 64, 0, stream>>>(hdr);
    hist_kernel<<<nb, 256, 0, stream>>>(idx, n, counts);
    scan_kernel<<<1, 32, 0, stream>>>(counts, offs, cursors);
    scatter_kernel<<<nb, 256, 0, stream>>>(idx, n, cursors, perm);

    dim3 grid2(16, 128);
    expert_mlp_wmma_kernel<<<grid2, 32, 0, stream>>>(
        x, W0, b0, W1, b1, W2, b2, offs, counts, perm, out);
}